// AddtiveAttention_8289286881976
// MI455X (gfx1250) — compile-verified
//
#include <hip/hip_runtime.h>
#include <math.h>

typedef __attribute__((ext_vector_type(2))) float v2f;
typedef __attribute__((ext_vector_type(8))) float v8f;

#define B_   8
#define LQ_  512
#define LK_  512
#define QF_  256
#define H_   128
#define VF_  256

// ---------------------------------------------------------------------------
// Kernel 1: fold the H=128 projection into the feature dimension.
//   u_q[f] = sum_h Wq[h,f] * w_s[h],  u_k[f] = sum_h Wk[h,f] * w_s[h]
//   c      = (bq + bk) . w_s + b_s
// ---------------------------------------------------------------------------
__global__ __launch_bounds__(256) void fold_kernel(
    const float* __restrict__ Wq, const float* __restrict__ bq,
    const float* __restrict__ Wk, const float* __restrict__ bk,
    const float* __restrict__ w_s, const float* __restrict__ b_s,
    float* __restrict__ u_q, float* __restrict__ u_k, float* __restrict__ cbias) {
  const int f = threadIdx.x;  // 0..255
  float sq = 0.f, sk = 0.f;
  #pragma unroll 8
  for (int h = 0; h < H_; ++h) {
    const float w = w_s[h];
    sq = fmaf(Wq[h * QF_ + f], w, sq);
    sk = fmaf(Wk[h * QF_ + f], w, sk);
  }
  u_q[f] = sq;
  u_k[f] = sk;
  if (f == 0) {
    float c = b_s[0];
    for (int h = 0; h < H_; ++h) c = fmaf(bq[h] + bk[h], w_s[h], c);
    cbias[0] = c;
  }
}

// ---------------------------------------------------------------------------
// Kernel 2: per-row dot products (GEMV), one wave32 per row.
//   rows [0, 4096)      -> qs[row]  = queries_row . u_q
//   rows [4096, 8192)   -> ks[row'] = keys_row    . u_k
// ---------------------------------------------------------------------------
__global__ __launch_bounds__(256) void rowdot_kernel(
    const float* __restrict__ queries, const float* __restrict__ keys,
    const float* __restrict__ u_q, const float* __restrict__ u_k,
    float* __restrict__ qs, float* __restrict__ ks) {
  const int wave = threadIdx.x >> 5;
  const int lane = threadIdx.x & 31;
  const int row  = blockIdx.x * 8 + wave;  // 0..8191

  const float* x; const float* u; float* outp; int r;
  if (row < B_ * LQ_) { x = queries; u = u_q; outp = qs; r = row; }
  else                { x = keys;    u = u_k; outp = ks; r = row - B_ * LQ_; }

  const float* xr = x + (size_t)r * QF_;
  float acc = 0.f;
  #pragma unroll
  for (int t = 0; t < QF_ / 32; ++t) {
    const int f = lane + 32 * t;
    acc = fmaf(xr[f], u[f], acc);
  }
  #pragma unroll
  for (int off = 16; off > 0; off >>= 1)
    acc += __shfl_xor(acc, off, 32);
  if (lane == 0) outp[r] = acc;
}

// ---------------------------------------------------------------------------
// Kernel 3: fused scores -> masked softmax -> attn @ V using f32 WMMA.
// One block (8 wave32) per (batch b, 16-row q-tile).
//   scores[i,j] = tanh(qs[i] + ks[j] + c), mask j >= valid_len -> -1e6
//   probs into LDS [16][516] (stride 516: conflict-free, 8B-aligned pairs)
//   each wave accumulates TWO 16x16 output tiles sharing one A fragment:
//   K-loop unrolled 4x -> 8 back-to-back V_WMMA_F32_16X16X4_F32 per iter.
// ---------------------------------------------------------------------------
__global__ __launch_bounds__(256) void attn_kernel(
    const float* __restrict__ values, const int* __restrict__ valid_lens,
    const float* __restrict__ qs, const float* __restrict__ ks,
    const float* __restrict__ cbias, float* __restrict__ out) {
  __shared__ float probs[16 * 516];

  const int b    = blockIdx.x >> 5;   // LQ_/16 = 32 q-tiles per batch
  const int qt   = blockIdx.x & 31;
  const int wave = threadIdx.x >> 5;
  const int lane = threadIdx.x & 31;
  const int vlen = valid_lens[b];
  const float c  = cbias[0];
  const float* ksb = ks + b * LK_;

  // ---- softmax phase: each wave owns 2 of the 16 q-rows ----
  #pragma unroll
  for (int rr = 0; rr < 2; ++rr) {
    const int m = wave * 2 + rr;
    const float qv = qs[b * LQ_ + qt * 16 + m] + c;

    float sc[16];
    float mx = -3.0e38f;
    #pragma unroll
    for (int t = 0; t < 16; ++t) {
      const int j = lane + 32 * t;
      float s = tanhf(qv + ksb[j]);
      if (j >= vlen) s = -1.0e6f;
      sc[t] = s;
      mx = fmaxf(mx, s);
    }
    #pragma unroll
    for (int off = 16; off > 0; off >>= 1)
      mx = fmaxf(mx, __shfl_xor(mx, off, 32));

    float sum = 0.f;
    #pragma unroll
    for (int t = 0; t < 16; ++t) {
      const float p = __expf(sc[t] - mx);  // masked entries underflow to 0.0f
      sc[t] = p;
      sum += p;
    }
    #pragma unroll
    for (int off = 16; off > 0; off >>= 1)
      sum += __shfl_xor(sum, off, 32);

    const float rinv = 1.0f / sum;
    #pragma unroll
    for (int t = 0; t < 16; ++t) {
      const int j = lane + 32 * t;
      probs[m * 516 + j] = sc[t] * rinv;
    }
  }
  __syncthreads();

  // ---- WMMA phase ----
  // A fragment (16x4 f32): lane holds M = lane&15, K = 2*(lane>>4)+{0,1}
  // B fragment (4x16 f32): lane holds N = lane&15, K = 2*(lane>>4)+{0,1}
  // D fragment: VGPR r -> row r + 8*(lane>>4), col lane&15
  const int arow  = lane & 15;
  const int kbase = (lane >> 4) * 2;
  // probs are exactly 0 for j >= vlen: round K up to a multiple of 16 so the
  // loop unrolls cleanly by 4 (vlen in [1,512] -> kmax in [16,512]).
  const int kmax  = ((vlen + 15) >> 4) << 4;
  const float* Vb = values + (size_t)b * LK_ * VF_;
  float* ob = out + (size_t)(b * LQ_ + qt * 16) * VF_;

  const int ncol0 = wave * 32 + arow;   // n-tile 2*wave
  const int ncol1 = ncol0 + 16;         // n-tile 2*wave + 1
  v8f acc0 = {0.f, 0.f, 0.f, 0.f, 0.f, 0.f, 0.f, 0.f};
  v8f acc1 = {0.f, 0.f, 0.f, 0.f, 0.f, 0.f, 0.f, 0.f};

  #pragma unroll 4
  for (int k = 0; k < kmax; k += 4) {
    v2f a, b0, b1;
    a.x = probs[arow * 516 + k + kbase];
    a.y = probs[arow * 516 + k + kbase + 1];
    const float* vrow0 = Vb + (size_t)(k + kbase) * VF_;
    const float* vrow1 = vrow0 + VF_;
    b0.x = vrow0[ncol0];
    b0.y = vrow1[ncol0];
    b1.x = vrow0[ncol1];
    b1.y = vrow1[ncol1];
    acc0 = __builtin_amdgcn_wmma_f32_16x16x4_f32(
        false, a, false, b0, (short)0, acc0, false, false);
    acc1 = __builtin_amdgcn_wmma_f32_16x16x4_f32(
        false, a, false, b1, (short)0, acc1, false, false);
  }

  const int orow_off = (lane >> 4) * 8;
  #pragma unroll
  for (int r = 0; r < 8; ++r) {
    const int row = r + orow_off;
    ob[(size_t)row * VF_ + ncol0] = acc0[r];
    ob[(size_t)row * VF_ + ncol1] = acc1[r];
  }
}

// ---------------------------------------------------------------------------
extern "C" void kernel_launch(void* const* d_in, const int* in_sizes, int n_in,
                              void* d_out, int out_size, void* d_ws, size_t ws_size,
                              hipStream_t stream) {
  const float* queries = (const float*)d_in[0];
  const float* keys    = (const float*)d_in[1];
  const float* values  = (const float*)d_in[2];
  const int*   vlens   = (const int*)  d_in[3];
  const float* Wq      = (const float*)d_in[4];
  const float* bq      = (const float*)d_in[5];
  const float* Wk      = (const float*)d_in[6];
  const float* bk      = (const float*)d_in[7];
  const float* w_s     = (const float*)d_in[8];
  const float* b_s     = (const float*)d_in[9];
  float* out = (float*)d_out;

  float* ws  = (float*)d_ws;
  float* u_q = ws;                 // 256
  float* u_k = ws + 256;           // 256
  float* cb  = ws + 512;           // 1 (padded to 4)
  float* qsv = ws + 516;           // B*LQ = 4096
  float* ksv = ws + 516 + 4096;    // B*LK = 4096

  fold_kernel<<<1, 256, 0, stream>>>(Wq, bq, Wk, bk, w_s, b_s, u_q, u_k, cb);
  rowdot_kernel<<<1024, 256, 0, stream>>>(queries, keys, u_q, u_k, qsv, ksv);
  attn_kernel<<<B_ * (LQ_ / 16), 256, 0, stream>>>(values, vlens, qsv, ksv, cb, out);
}